// Net_6622839570835
// MI455X (gfx1250) — compile-verified
//
#include <hip/hip_runtime.h>

typedef __attribute__((ext_vector_type(2))) float v2f;
typedef __attribute__((ext_vector_type(8))) float v8f;

#define F_IN 512
#define F_H  128
#define F_OUT 64
#define PN_EPS 1e-5f

// ---------------- utility ----------------
__global__ void zero_f32(float* __restrict__ p, long long n) {
    long long i = (long long)blockIdx.x * blockDim.x + threadIdx.x;
    long long stride = (long long)gridDim.x * blockDim.x;
    for (; i < n; i += stride) p[i] = 0.0f;
}

__global__ void init_out_kernel(float* __restrict__ out, const float* __restrict__ b2, long long n) {
    long long i = (long long)blockIdx.x * blockDim.x + threadIdx.x;
    long long stride = (long long)gridDim.x * blockDim.x;
    for (; i < n; i += stride) out[i] = b2[i & (F_OUT - 1)];
}

// ---------------- GCN norm ----------------
__global__ void deg_kernel(const int* __restrict__ col, int E, float* __restrict__ deg) {
    int i = blockIdx.x * blockDim.x + threadIdx.x;
    if (i < E) atomicAdd(&deg[col[i]], 1.0f);
}

__global__ void dinv_kernel(float* __restrict__ deg, int N) {
    int i = blockIdx.x * blockDim.x + threadIdx.x;
    if (i < N) deg[i] = rsqrtf(deg[i] + 1.0f);   // +1 self loop; deg>0 always
}

// ---------------- PairNorm stats ----------------
__global__ void colstats_kernel(const float* __restrict__ x, int N,
                                float* __restrict__ colsum, float* __restrict__ colsumsq) {
    int c = threadIdx.x;                         // 512 threads: one column each
    float s = 0.0f, sq = 0.0f;
    for (int n = blockIdx.x; n < N; n += gridDim.x) {
        float v = x[(long long)n * F_IN + c];    // coalesced: 512 consecutive floats/row
        s += v; sq += v * v;
    }
    atomicAdd(&colsum[c], s);
    atomicAdd(&colsumsq[c], sq);
}

__global__ void stats_finalize(const float* __restrict__ colsum, const float* __restrict__ colsumsq,
                               float* __restrict__ mean, float* __restrict__ invden, int N) {
    __shared__ float red[F_IN];
    int c = threadIdx.x;
    float invN = 1.0f / (float)N;
    float m = colsum[c] * invN;
    mean[c] = m;
    red[c] = colsumsq[c] - (float)N * m * m;     // sum_n (x-mean)^2 for this column
    __syncthreads();
    for (int s = F_IN / 2; s > 0; s >>= 1) {
        if (c < s) red[c] += red[c + s];
        __syncthreads();
    }
    if (c == 0) invden[0] = rsqrtf(PN_EPS + red[0] * invN);
}

__global__ void meanW1_kernel(const float* __restrict__ mean, const float* __restrict__ W1,
                              float* __restrict__ mW1) {
    int k = threadIdx.x;                         // 128 threads
    float s = 0.0f;
    for (int c = 0; c < F_IN; ++c) s += mean[c] * W1[c * F_H + k];
    mW1[k] = s;
}

// ---------------- GEMM1: h1 = ((x - mean) * invden) @ W1 (fused) ----------------
// Block: 256 threads = 8 waves; 16 rows x 128 cols per block; K=512 in LDS.
#define AS1 (F_IN + 4)   // pad: 516 % 64 == 4 -> rows hit distinct LDS banks
__global__ void __launch_bounds__(256)
gemm1_wmma(const float* __restrict__ x, const float* __restrict__ W1,
           const float* __restrict__ mW1, const float* __restrict__ invden,
           float* __restrict__ h1, int N) {
    __shared__ float As[16 * AS1];
    int tid = threadIdx.x;
    long long rowBase = (long long)blockIdx.x * 16;

    for (int i = tid; i < 16 * F_IN; i += 256) {
        int r = i / F_IN, k = i - r * F_IN;
        long long gr = rowBase + r; if (gr >= N) gr = N - 1;
        As[r * AS1 + k] = x[gr * F_IN + k];
    }
    __syncthreads();

    int wave = tid >> 5, lane = tid & 31;
    int half = lane >> 4, mn = lane & 15;
    int cb = wave * 16;                          // column tile for this wave
    v8f acc = {};
    for (int k0 = 0; k0 < F_IN; k0 += 4) {
        int ka = k0 + 2 * half;
        v2f a, b;
        a.x = As[mn * AS1 + ka];
        a.y = As[mn * AS1 + ka + 1];
        b.x = W1[(long long)ka * F_H + cb + mn];
        b.y = W1[(long long)(ka + 1) * F_H + cb + mn];
        acc = __builtin_amdgcn_wmma_f32_16x16x4_f32(false, a, false, b,
                                                    (short)0, acc, false, false);
    }
    float inv = invden[0];
    int col = cb + mn;
    float mw = mW1[col];
    for (int v = 0; v < 8; ++v) {
        long long r = rowBase + v + 8 * half;
        if (r < N) h1[r * F_H + col] = (acc[v] - mw) * inv;
    }
}

// ---------------- scatter layer 1: out1[col] += h1[row] * ew ----------------
__global__ void scatter1_kernel(const float* __restrict__ h1, const int* __restrict__ row,
                                const int* __restrict__ col, const float* __restrict__ dinv,
                                float* __restrict__ out1, int E, int N) {
    long long tid = (long long)blockIdx.x * blockDim.x + threadIdx.x;
    long long e = tid >> 5; int lane = (int)(tid & 31);   // 32 lanes x float4 = 128 feats
    if (e >= (long long)E + N) return;
    int r, c;
    if (e < E) { r = row[e]; c = col[e]; } else { r = c = (int)(e - E); }
    float w = dinv[r] * dinv[c];
    float4 v = ((const float4*)(h1 + (long long)r * F_H))[lane];
    float* dst = out1 + (long long)c * F_H + lane * 4;
    atomicAdd(dst + 0, v.x * w);
    atomicAdd(dst + 1, v.y * w);
    atomicAdd(dst + 2, v.z * w);
    atomicAdd(dst + 3, v.w * w);
}

// ---------------- GEMM2: h2 = relu(out1 + b1) @ W2 (bias+relu fused in A load) ----------------
#define AS2 (F_H + 4)    // 132 % 64 == 4 -> conflict-free
__global__ void __launch_bounds__(128)
gemm2_wmma(const float* __restrict__ out1, const float* __restrict__ b1,
           const float* __restrict__ W2, float* __restrict__ h2, int N) {
    __shared__ float As[16 * AS2];
    int tid = threadIdx.x;
    long long rowBase = (long long)blockIdx.x * 16;

    for (int i = tid; i < 16 * F_H; i += 128) {
        int r = i / F_H, k = i - r * F_H;
        long long gr = rowBase + r; if (gr >= N) gr = N - 1;
        float v = out1[gr * F_H + k] + b1[k];
        As[r * AS2 + k] = v > 0.0f ? v : 0.0f;
    }
    __syncthreads();

    int wave = tid >> 5, lane = tid & 31;       // 4 waves x 16 cols = 64 output cols
    int half = lane >> 4, mn = lane & 15;
    int cb = wave * 16;
    v8f acc = {};
    for (int k0 = 0; k0 < F_H; k0 += 4) {
        int ka = k0 + 2 * half;
        v2f a, b;
        a.x = As[mn * AS2 + ka];
        a.y = As[mn * AS2 + ka + 1];
        b.x = W2[ka * F_OUT + cb + mn];
        b.y = W2[(ka + 1) * F_OUT + cb + mn];
        acc = __builtin_amdgcn_wmma_f32_16x16x4_f32(false, a, false, b,
                                                    (short)0, acc, false, false);
    }
    int col = cb + mn;
    for (int v = 0; v < 8; ++v) {
        long long r = rowBase + v + 8 * half;
        if (r < N) h2[r * F_OUT + col] = acc[v];
    }
}

// ---------------- scatter layer 2: out[col] += h2[row] * ew (out pre-filled with b2) ----------------
__global__ void scatter2_kernel(const float* __restrict__ h2, const int* __restrict__ row,
                                const int* __restrict__ col, const float* __restrict__ dinv,
                                float* __restrict__ out, int E, int N) {
    long long tid = (long long)blockIdx.x * blockDim.x + threadIdx.x;
    long long e = tid >> 4; int lane = (int)(tid & 15);   // 16 lanes x float4 = 64 feats
    if (e >= (long long)E + N) return;
    int r, c;
    if (e < E) { r = row[e]; c = col[e]; } else { r = c = (int)(e - E); }
    float w = dinv[r] * dinv[c];
    float4 v = ((const float4*)(h2 + (long long)r * F_OUT))[lane];
    float* dst = out + (long long)c * F_OUT + lane * 4;
    atomicAdd(dst + 0, v.x * w);
    atomicAdd(dst + 1, v.y * w);
    atomicAdd(dst + 2, v.z * w);
    atomicAdd(dst + 3, v.w * w);
}

// ---------------- launch ----------------
extern "C" void kernel_launch(void* const* d_in, const int* in_sizes, int n_in,
                              void* d_out, int out_size, void* d_ws, size_t ws_size,
                              hipStream_t stream) {
    const float* x  = (const float*)d_in[0];
    const int*   ei = (const int*)d_in[1];
    const float* W1 = (const float*)d_in[2];
    const float* b1 = (const float*)d_in[3];
    const float* W2 = (const float*)d_in[4];
    const float* b2 = (const float*)d_in[5];
    float* out = (float*)d_out;

    int N = in_sizes[0] / F_IN;
    int E = in_sizes[1] / 2;
    const int* rowp = ei;       // edge_index[0]
    const int* colp = ei + E;   // edge_index[1]

    // workspace layout (floats)
    float* ws       = (float*)d_ws;
    float* deg      = ws;                              // N (becomes dinv in place)
    float* colsum   = deg + N;                         // 512
    float* colsumsq = colsum + F_IN;                   // 512
    float* mean     = colsumsq + F_IN;                 // 512
    float* invden   = mean + F_IN;                     // 16
    float* mW1      = invden + 16;                     // 128
    float* h1       = mW1 + F_H;                       // N*128
    float* out1     = h1 + (long long)N * F_H;         // N*128
    float* h2       = out1 + (long long)N * F_H;       // N*64

    long long statsN = (long long)N + 3 * F_IN + 16 + F_H;
    zero_f32<<<1024, 256, 0, stream>>>(ws, statsN);
    zero_f32<<<4096, 256, 0, stream>>>(out1, (long long)N * F_H);

    deg_kernel<<<(E + 255) / 256, 256, 0, stream>>>(colp, E, deg);
    dinv_kernel<<<(N + 255) / 256, 256, 0, stream>>>(deg, N);

    colstats_kernel<<<512, F_IN, 0, stream>>>(x, N, colsum, colsumsq);
    stats_finalize<<<1, F_IN, 0, stream>>>(colsum, colsumsq, mean, invden, N);
    meanW1_kernel<<<1, F_H, 0, stream>>>(mean, W1, mW1);

    gemm1_wmma<<<(N + 15) / 16, 256, 0, stream>>>(x, W1, mW1, invden, h1, N);

    long long work1 = ((long long)E + N) * 32;
    scatter1_kernel<<<(unsigned)((work1 + 255) / 256), 256, 0, stream>>>(h1, rowp, colp, deg, out1, E, N);

    gemm2_wmma<<<(N + 15) / 16, 128, 0, stream>>>(out1, b1, W2, h2, N);

    init_out_kernel<<<2048, 256, 0, stream>>>(out, b2, (long long)N * F_OUT);

    long long work2 = ((long long)E + N) * 16;
    scatter2_kernel<<<(unsigned)((work2 + 255) / 256), 256, 0, stream>>>(h2, rowp, colp, deg, out, E, N);
}